// ScaledDotProductAttention_266
// MI455X (gfx1250) — compile-verified
//
#include <hip/hip_runtime.h>
#include <cstddef>

// Scaled dot-product attention for MI455X (gfx1250, wave32).
// out[B,H,S,D] and attn[B,H,S,S] both fp32, concatenated flat in d_out.
// B=4 H=16 S=2048 D=64. attn write (1.07 GB) dominates the roofline, so we
// keep everything fp32 and use V_WMMA_F32_16X16X4_F32 for both GEMMs.

typedef __attribute__((ext_vector_type(2))) float v2f;
typedef __attribute__((ext_vector_type(8))) float v8f;

#define B_ 4
#define H_ 16
#define S_ 2048
#define D_ 64
#define QT 16              // query rows per workgroup
#define NKT (S_ / 16)      // 128 key tiles of 16
#define LDSROW (S_ + 4)    // +4 floats pad -> bank-conflict-free strided loads
#define NEG_INF (-1000000000.0f)

__launch_bounds__(128)
__global__ void sdpa_kernel(const float* __restrict__ qg,
                            const float* __restrict__ kg,
                            const float* __restrict__ vg,
                            const int*   __restrict__ mg,
                            float* __restrict__ outg,
                            float* __restrict__ attng)
{
    extern __shared__ float smem[];
    float* sS   = smem;                      // [QT][LDSROW] scores -> probs
    float* sRed = smem + QT * LDSROW;        // [QT][8] partial reductions
    float* sMax = sRed + QT * 8;             // [QT] row max
    float* sInv = sMax + QT;                 // [QT] 1/rowsum

    const int tid  = threadIdx.x;
    const int wave = tid >> 5;
    const int lane = tid & 31;
    // WMMA f32 16x16x4 lane decomposition (ISA 7.12.2):
    //   A: lanes 0-15 / 16-31 both hold M=0..15; VGPR pair = K {0,1} / {2,3}
    //   C/D: VGPR g holds M=g (lanes 0-15) and M=g+8 (lanes 16-31), N=lane&15
    const int am  = lane & 15;          // M for A, N for B/C/D
    const int ak  = (lane >> 4) << 1;   // K sub-offset: 0 or 2
    const int rhi = (lane >> 4) << 3;   // C/D row offset: 0 or 8

    const int bh = blockIdx.y;          // fused B*H, 0..63
    const int b  = bh >> 4;             // H_ = 16
    const int q0 = blockIdx.x * QT;

    const size_t head_off = (size_t)bh * S_ * D_;
    const float* Q = qg + head_off + (size_t)q0 * D_;
    const float* K = kg + head_off;
    const float* V = vg + head_off;
    const int*   M = mg + ((size_t)b * S_ + q0) * S_;
    float* OUT  = outg  + head_off + (size_t)q0 * D_;
    float* ATTN = attng + ((size_t)bh * S_ + q0) * S_;

    // Cache the 16x64 Q tile as 16 A-fragments (32 VGPRs), reused for all 32
    // key tiles this wave touches.
    v2f aq[D_ / 4];
#pragma unroll
    for (int c = 0; c < D_ / 4; ++c) {
        const float* p = Q + (size_t)am * D_ + c * 4 + ak;
        aq[c] = {p[0], p[1]};
    }

    // ---- Phase 1: S = scale * Q K^T, masked, into LDS -------------------
    for (int kt = wave; kt < NKT; kt += 4) {
        if (kt + 4 < NKT)
            __builtin_prefetch(K + (size_t)((kt + 4) * 16 + am) * D_, 0, 0);
        const float* kp = K + (size_t)(kt * 16 + am) * D_ + ak;
        v8f acc = {};
#pragma unroll
        for (int c = 0; c < D_ / 4; ++c) {
            v2f bk = {kp[c * 4], kp[c * 4 + 1]};   // B[kk][n] = K[n][d]
            acc = __builtin_amdgcn_wmma_f32_16x16x4_f32(
                false, aq[c], false, bk, (short)0, acc, false, false);
        }
        const int kn = kt * 16 + am;               // key/column index
#pragma unroll
        for (int g = 0; g < 8; ++g) {
            const int m = g + rhi;
            float s = acc[g] * 0.125f;             // 1/sqrt(64)
            s = (M[(size_t)m * S_ + kn] == 0) ? NEG_INF : s;
            sS[m * LDSROW + kn] = s;
        }
    }
    __syncthreads();

    // ---- Phase 2: row max and sum(exp) over S=2048 ----------------------
    const int rr = tid >> 3;   // row 0..15
    const int sg = tid & 7;    // segment 0..7 (stride-8 scan: bank friendly)
    float mx = NEG_INF;
    for (int t = 0; t < S_ / 8; ++t)
        mx = fmaxf(mx, sS[rr * LDSROW + sg + 8 * t]);
    sRed[rr * 8 + sg] = mx;
    __syncthreads();
    if (sg == 0) {
        float m2 = sRed[rr * 8];
#pragma unroll
        for (int j = 1; j < 8; ++j) m2 = fmaxf(m2, sRed[rr * 8 + j]);
        sMax[rr] = m2;
    }
    __syncthreads();
    const float rowm = sMax[rr];
    float sum = 0.0f;
    for (int t = 0; t < S_ / 8; ++t)
        sum += __expf(sS[rr * LDSROW + sg + 8 * t] - rowm);
    sRed[rr * 8 + sg] = sum;
    __syncthreads();
    if (sg == 0) {
        float s2 = 0.0f;
#pragma unroll
        for (int j = 0; j < 8; ++j) s2 += sRed[rr * 8 + j];
        sInv[rr] = 1.0f / s2;    // fully-masked row: max=NEG_INF -> sum=2048
    }
    __syncthreads();

    // ---- Phase 3: P in LDS + streamed (non-temporal) attn write ---------
#pragma unroll 1
    for (int r = 0; r < QT; ++r) {
        const float rm = sMax[r];
        const float ri = sInv[r];
        for (int j = tid; j < S_; j += 128) {      // coalesced per row
            const float p = __expf(sS[r * LDSROW + j] - rm) * ri;
            sS[r * LDSROW + j] = p;
            __builtin_nontemporal_store(p, ATTN + (size_t)r * S_ + j);
        }
    }
    __syncthreads();

    // ---- Phase 4: out = P(16x2048) @ V(2048x64); wave w -> cols [16w,16w+16)
    {
        const int n0 = wave * 16;
        const float* vp = V + n0 + am;             // column n0+am of V
        v8f acc = {};
#pragma unroll 4
        for (int c = 0; c < S_ / 4; ++c) {
            const int kb = c * 4 + ak;
            v2f ap = {sS[am * LDSROW + kb], sS[am * LDSROW + kb + 1]};
            v2f bp = {vp[(size_t)kb * D_], vp[(size_t)(kb + 1) * D_]};
            acc = __builtin_amdgcn_wmma_f32_16x16x4_f32(
                false, ap, false, bp, (short)0, acc, false, false);
        }
#pragma unroll
        for (int g = 0; g < 8; ++g)
            OUT[(size_t)(g + rhi) * D_ + n0 + am] = acc[g];
    }
}

extern "C" void kernel_launch(void* const* d_in, const int* in_sizes, int n_in,
                              void* d_out, int out_size, void* d_ws, size_t ws_size,
                              hipStream_t stream) {
    (void)in_sizes; (void)n_in; (void)out_size; (void)d_ws; (void)ws_size;
    const float* q = (const float*)d_in[0];
    const float* k = (const float*)d_in[1];
    const float* v = (const float*)d_in[2];
    const int*   m = (const int*)d_in[3];
    float* out  = (float*)d_out;                       // [B,H,S,D]
    float* attn = out + (size_t)B_ * H_ * S_ * D_;     // [B,H,S,S]

    const size_t shmem = (size_t)(QT * LDSROW + QT * 8 + QT + QT) * sizeof(float);
    dim3 grid(S_ / QT, B_ * H_);   // 128 query tiles x 64 heads
    sdpa_kernel<<<grid, 128, shmem, stream>>>(q, k, v, m, out, attn);
}